// Canvas_DIP_by_distance_70196945485842
// MI455X (gfx1250) — compile-verified
//
#include <hip/hip_runtime.h>
#include <hip/hip_bf16.h>

// ---------------------------------------------------------------------------
// Canvas_DIP_by_distance on MI455X (gfx1250, wave32)
//
// Forward math: out_color(h,w) = palette[ argmax_c ||palette[c]-sigmoid(wl[h,w])||^2 ]
// then nearest-upsample (3,224,224) -> (3,2048,2048).
//
// Stage 1 uses V_WMMA_F32_16X16X4_F32 for the -2*P.W dot-product term of the
// squared distance (K=3 padded to 4 with zeros), one wave per 16-pixel tile,
// 4 WMMAs sweep all 64 palette colors. Stage 2 is the HBM-bound upsample.
// ---------------------------------------------------------------------------

typedef __attribute__((ext_vector_type(2))) float v2f;
typedef __attribute__((ext_vector_type(8))) float v8f;

#define CANVAS_H 224
#define CANVAS_W 224
#define NPIX     (CANVAS_H * CANVAS_W)   // 50176 = 3136 * 16
#define NCOL     64
#define IMG_H    2048
#define IMG_W    2048

__device__ __forceinline__ float sigmoidf_fast(float x) {
    return 1.0f / (1.0f + __expf(-x));
}

// One wave32 handles a 16-pixel M-tile against all 64 palette colors.
// 8 waves / block, 392 blocks -> exactly 3136 tiles, no tail (EXEC all-1s,
// required for WMMA).
__global__ __launch_bounds__(256) void canvas_argmax_wmma(
    const float* __restrict__ wl,    // (NPIX, 3) weight logits
    const float* __restrict__ pal,   // (NCOL, 3) palette
    int*         __restrict__ idxOut)// (NPIX) winning palette index
{
    const int lane  = threadIdx.x & 31;
    const int wave  = (blockIdx.x << 3) + (threadIdx.x >> 5);
    const int m0    = wave << 4;          // first pixel of this 16-row tile
    const int half  = lane >> 4;          // 0: lanes 0-15, 1: lanes 16-31
    const int lm    = lane & 15;

    // ---- Build A (16x4, f32): row M = lm. ISA layout:
    //   lanes 0-15 : VGPR0=K0, VGPR1=K1 ; lanes 16-31 : VGPR0=K2, VGPR1=K3(pad)
    const int m = m0 + lm;
    const float w0 = sigmoidf_fast(wl[m * 3 + 0]);
    const float w1 = sigmoidf_fast(wl[m * 3 + 1]);
    const float w2 = sigmoidf_fast(wl[m * 3 + 2]);
    v2f a;
    a.x = half ? w2   : w0;
    a.y = half ? 0.0f : w1;

    float bestScore[8];
    int   bestN[8];
#pragma unroll
    for (int i = 0; i < 8; ++i) { bestScore[i] = -3.0e38f; bestN[i] = 0x7fffffff; }

    // ---- Sweep the 64 palette colors as 4 N-tiles of 16
#pragma unroll
    for (int t = 0; t < 4; ++t) {
        const int n = (t << 4) + lm;      // this lane's palette column
        const float p0 = pal[n * 3 + 0];
        const float p1 = pal[n * 3 + 1];
        const float p2 = pal[n * 3 + 2];
        const float pn = p0 * p0 + p1 * p1 + p2 * p2;   // ||p_n||^2

        // B (4x16, f32): column N = lm; mirror of A layout across K halves.
        v2f b;
        b.x = half ? p2   : p0;
        b.y = half ? 0.0f : p1;

        v8f c = {};
        c = __builtin_amdgcn_wmma_f32_16x16x4_f32(
                /*neg_a=*/false, a, /*neg_b=*/false, b,
                /*c_mod=*/(short)0, c, /*reuse_a=*/false, /*reuse_b=*/false);

        // D layout: lane holds column N=lm, rows M = i + 8*half for VGPR i.
        // score = ||p||^2 - 2 * dot(w_m, p_n)  (||w||^2 constant per row)
#pragma unroll
        for (int i = 0; i < 8; ++i) {
            const float s = pn - 2.0f * c[i];
            if (s > bestScore[i] || (s == bestScore[i] && n < bestN[i])) {
                bestScore[i] = s;
                bestN[i]     = n;
            }
        }
    }

    // ---- Argmax across the 16 lanes of each half (butterfly, wave32-native)
#pragma unroll
    for (int off = 8; off >= 1; off >>= 1) {
#pragma unroll
        for (int i = 0; i < 8; ++i) {
            const float os = __shfl_xor(bestScore[i], off, 16);
            const int   on = __shfl_xor(bestN[i],     off, 16);
            if (os > bestScore[i] || (os == bestScore[i] && on < bestN[i])) {
                bestScore[i] = os;
                bestN[i]     = on;
            }
        }
    }

    // Lane (half, lm<8) writes row half*8 + lm using bestN[lm]
    if (lm < 8) {
        int sel = bestN[0];
#pragma unroll
        for (int i = 1; i < 8; ++i) {
            if (lm == i) sel = bestN[i];
        }
        idxOut[m0 + (half << 3) + lm] = sel;
    }
}

// Nearest-upsample + palette gather; pure HBM-store-bound (~50 MB out).
__global__ __launch_bounds__(256) void upsample_lookup(
    const int*   __restrict__ idx,   // (NPIX)
    const float* __restrict__ pal,   // (NCOL, 3)
    float*       __restrict__ out)   // (3, IMG_H, IMG_W)
{
    const int tid   = blockIdx.x * 256 + threadIdx.x;
    const int total = IMG_H * IMG_W;
    if (tid >= total) return;

    const int y  = tid >> 11;                 // / IMG_W
    const int x  = tid & (IMG_W - 1);
    const int ih = (y * CANVAS_H) / IMG_H;    // floor(y*224/2048) = (y*7)>>6
    const int iw = (x * CANVAS_W) / IMG_W;

    const int   c = idx[ih * CANVAS_W + iw];
    const float r = pal[c * 3 + 0];
    const float g = pal[c * 3 + 1];
    const float b = pal[c * 3 + 2];

    out[tid]             = r;
    out[total + tid]     = g;
    out[2 * total + tid] = b;
}

extern "C" void kernel_launch(void* const* d_in, const int* in_sizes, int n_in,
                              void* d_out, int out_size, void* d_ws, size_t ws_size,
                              hipStream_t stream) {
    const float* wl  = (const float*)d_in[0];   // (224,224,3) f32
    const float* pal = (const float*)d_in[1];   // (64,3) f32
    float*       out = (float*)d_out;           // (3,2048,2048) f32
    int*         idx = (int*)d_ws;              // 50176 ints of scratch

    // 3136 M-tiles, 8 waves/block -> 392 blocks, exact cover (no tail).
    canvas_argmax_wmma<<<NPIX / 16 / 8, 256, 0, stream>>>(wl, pal, idx);

    const int total = IMG_H * IMG_W;
    upsample_lookup<<<(total + 255) / 256, 256, 0, stream>>>(idx, pal, out);
}